// DMPNN_78065325572490
// MI455X (gfx1250) — compile-verified
//
#include <hip/hip_runtime.h>
#include <math.h>

#define NN      100000   // nodes
#define NE      400000   // edges
#define NLG     1200000  // line-graph edges
#define NG      2000     // graphs
#define EDIM    16
#define NF      64
#define NITER   4

typedef __attribute__((ext_vector_type(2))) float v2f;
typedef __attribute__((ext_vector_type(8))) float v8f;

enum { ACT_NONE = 0, ACT_LRELU = 1, ACT_TANH = 2 };

__device__ __forceinline__ float apply_act(float x, int act) {
    if (act == ACT_LRELU) return x > 0.f ? x : 0.01f * x;
    if (act == ACT_TANH)  return tanhf(x);
    return x;
}

// ---------------------------------------------------------------------------
// C[M,64] = act(A[M,K] @ W[K,64] + bias)  via V_WMMA_F32_16X16X4_F32.
// One wave (32 lanes) per 16-row strip; 4 N-tiles of 16 cols each.
// A 16x4 f32 frag: lanes 0-15 -> M=lane, {K0,K1}; lanes 16-31 -> M=lane-16, {K2,K3}.
// B 4x16 f32 frag: lanes 0-15 -> N=lane, {K0,K1}; lanes 16-31 -> N=lane-16, {K2,K3}.
// C 16x16 f32: vgpr v, lane<16 -> (M=v,   N=lane); lane>=16 -> (M=v+8, N=lane-16).
// M must be a multiple of 16 (100000, 400000, 2000 all are).
// ---------------------------------------------------------------------------
__global__ __launch_bounds__(32)
void gemm_wmma_f32(const float* __restrict__ A, int lda, int K,
                   const float* __restrict__ W,
                   const float* __restrict__ bias,
                   float* __restrict__ C, int act)
{
    const int lane = threadIdx.x & 31;
    const int half = lane >> 4;     // 0 or 1
    const int lr   = lane & 15;
    const long row0 = (long)blockIdx.x * 16;

    v8f acc0 = {}, acc1 = {}, acc2 = {}, acc3 = {};

    for (int k0 = 0; k0 < K; k0 += 4) {
        const int ka = k0 + 2 * half;
        v2f a;
        const float* ap = A + (row0 + lr) * (long)lda + ka;
        a.x = ap[0];
        a.y = ap[1];

        const float* w0 = W + (size_t)ka * NF;       // row K=ka
        const float* w1 = w0 + NF;                   // row K=ka+1
        v2f b0, b1, b2, b3;
        b0.x = w0[lr +  0]; b0.y = w1[lr +  0];
        b1.x = w0[lr + 16]; b1.y = w1[lr + 16];
        b2.x = w0[lr + 32]; b2.y = w1[lr + 32];
        b3.x = w0[lr + 48]; b3.y = w1[lr + 48];

        acc0 = __builtin_amdgcn_wmma_f32_16x16x4_f32(false, a, false, b0, (short)0, acc0, false, false);
        acc1 = __builtin_amdgcn_wmma_f32_16x16x4_f32(false, a, false, b1, (short)0, acc1, false, false);
        acc2 = __builtin_amdgcn_wmma_f32_16x16x4_f32(false, a, false, b2, (short)0, acc2, false, false);
        acc3 = __builtin_amdgcn_wmma_f32_16x16x4_f32(false, a, false, b3, (short)0, acc3, false, false);
    }

    #pragma unroll
    for (int v = 0; v < 8; v++) {
        const int  row  = v + 8 * half;
        float*     crow = C + (row0 + row) * (long)NF;
        const int  c0 = lr, c1 = lr + 16, c2 = lr + 32, c3 = lr + 48;
        float b0 = bias ? bias[c0] : 0.f;
        float b1 = bias ? bias[c1] : 0.f;
        float b2 = bias ? bias[c2] : 0.f;
        float b3 = bias ? bias[c3] : 0.f;
        crow[c0] = apply_act(acc0[v] + b0, act);
        crow[c1] = apply_act(acc1[v] + b1, act);
        crow[c2] = apply_act(acc2[v] + b2, act);
        crow[c3] = apply_act(acc3[v] + b3, act);
    }
}

// e0[e,f] = (edge_u[src[e],f] + edge_v[dst[e],f] + e0[e,f]) / 3
__global__ void e0_combine(const float* __restrict__ edge_u,
                           const float* __restrict__ edge_v,
                           const int* __restrict__ src,
                           const int* __restrict__ dst,
                           float* __restrict__ e0)
{
    long idx = (long)blockIdx.x * blockDim.x + threadIdx.x;
    if (idx >= (long)NE * NF) return;
    const int e = (int)(idx >> 6);
    const int f = (int)(idx & 63);
    const float u = edge_u[(size_t)src[e] * NF + f];
    const float v = edge_v[(size_t)dst[e] * NF + f];
    e0[idx] = (u + v + e0[idx]) * (1.f / 3.f);
}

__global__ void copy_f(float* __restrict__ dst, const float* __restrict__ src, long n)
{
    long idx = (long)blockIdx.x * blockDim.x + threadIdx.x;
    if (idx < n) dst[idx] = src[idx];
}

// dst[ldst[i], f] += src[lsrc[i], f]  (feature-wide scatter over line graph)
__global__ void lg_scatter_feat(const int* __restrict__ lsrc,
                                const int* __restrict__ ldst,
                                const float* __restrict__ src,
                                float* __restrict__ dst)
{
    long idx = (long)blockIdx.x * blockDim.x + threadIdx.x;
    if (idx >= (long)NLG * NF) return;
    const int i = (int)(idx >> 6);
    const int f = (int)(idx & 63);
    atomicAdd(dst + (size_t)ldst[i] * NF + f, src[(size_t)lsrc[i] * NF + f]);
}

// r[e] = out[e]·conv_rel_w ; xconv[e] = out[e]·conv_root_w + conv_rel_b
// one wave per edge, butterfly reduce
__global__ __launch_bounds__(256)
void edge_dots(const float* __restrict__ out,
               const float* __restrict__ rel_w,
               const float* __restrict__ root_w,
               const float* __restrict__ rel_b,
               float* __restrict__ rbuf,
               float* __restrict__ xconv)
{
    const int lane = threadIdx.x & 31;
    const int wav  = threadIdx.x >> 5;
    const long e   = (long)blockIdx.x * 8 + wav;
    if (e >= NE) return;
    const float o0 = out[e * NF + lane];
    const float o1 = out[e * NF + lane + 32];
    float r  = o0 * rel_w[lane]  + o1 * rel_w[lane + 32];
    float xr = o0 * root_w[lane] + o1 * root_w[lane + 32];
    #pragma unroll
    for (int off = 16; off > 0; off >>= 1) {
        r  += __shfl_xor(r,  off, 32);
        xr += __shfl_xor(xr, off, 32);
    }
    if (lane == 0) {
        rbuf[e]  = r;
        xconv[e] = xr + rel_b[0];
    }
}

// xconv[ldst[i]] += r[lsrc[i]]  (scalar scatter — linearity trick for agg@conv_rel_w)
__global__ void lg_scatter_scalar(const int* __restrict__ lsrc,
                                  const int* __restrict__ ldst,
                                  const float* __restrict__ r,
                                  float* __restrict__ xconv)
{
    long i = (long)blockIdx.x * blockDim.x + threadIdx.x;
    if (i >= NLG) return;
    atomicAdd(xconv + ldst[i], r[lsrc[i]]);
}

__device__ __forceinline__ int lower_bound_i(const int* __restrict__ a, int n, int key)
{
    int lo = 0, hi = n;
    while (lo < hi) {
        int mid = (lo + hi) >> 1;
        if (a[mid] < key) lo = mid + 1; else hi = mid;
    }
    return lo;
}

// Per graph: segment softmax of xconv over sorted batch ids, then
// gx[g,f] = sum_e att[e] * out[e,f]
__global__ __launch_bounds__(256)
void graph_pool(const float* __restrict__ xconv,
                const int* __restrict__ batch,
                const float* __restrict__ out,
                float* __restrict__ gx)
{
    __shared__ float sred[256];
    const int g   = blockIdx.x;
    const int tid = threadIdx.x;
    const int lo  = lower_bound_i(batch, NE, g);
    const int hi  = lower_bound_i(batch, NE, g + 1);

    // segment max
    float m = -INFINITY;
    for (int e = lo + tid; e < hi; e += 256) m = fmaxf(m, xconv[e]);
    sred[tid] = m; __syncthreads();
    for (int s = 128; s > 0; s >>= 1) {
        if (tid < s) sred[tid] = fmaxf(sred[tid], sred[tid + s]);
        __syncthreads();
    }
    m = sred[0]; __syncthreads();

    // exp-sum
    float d = 0.f;
    for (int e = lo + tid; e < hi; e += 256) d += expf(xconv[e] - m);
    sred[tid] = d; __syncthreads();
    for (int s = 128; s > 0; s >>= 1) {
        if (tid < s) sred[tid] += sred[tid + s];
        __syncthreads();
    }
    d = sred[0]; __syncthreads();
    const float inv_d = (d > 0.f) ? 1.f / d : 0.f;

    // weighted feature pool: tid = chunk*64 + f
    const int f     = tid & 63;
    const int chunk = tid >> 6;   // 0..3
    float acc = 0.f;
    for (int e = lo + chunk; e < hi; e += 4) {
        const float a = expf(xconv[e] - m) * inv_d;
        acc += a * out[(size_t)e * NF + f];
    }
    sred[tid] = acc; __syncthreads();
    if (tid < 64)
        gx[(size_t)g * NF + f] = sred[f] + sred[f + 64] + sred[f + 128] + sred[f + 192];
}

// scores[b,t] = softmax_t( (gout_all[t][b,:]·att_w) * NF^-0.5 )
__global__ void scores_kernel(const float* __restrict__ gout_all,
                              const float* __restrict__ att_w,
                              float* __restrict__ scores)
{
    const int b = blockIdx.x * blockDim.x + threadIdx.x;
    if (b >= NG) return;
    const float scale = rsqrtf((float)NF);
    float s[NITER];
    float m = -INFINITY;
    #pragma unroll
    for (int t = 0; t < NITER; t++) {
        float acc = 0.f;
        const float* gp = gout_all + (size_t)t * NG * NF + (size_t)b * NF;
        for (int f = 0; f < NF; f++) acc += gp[f] * att_w[f];
        s[t] = acc * scale;
        m = fmaxf(m, s[t]);
    }
    float d = 0.f;
    #pragma unroll
    for (int t = 0; t < NITER; t++) { s[t] = expf(s[t] - m); d += s[t]; }
    const float inv_d = 1.f / d;
    #pragma unroll
    for (int t = 0; t < NITER; t++) scores[(size_t)b * NITER + t] = s[t] * inv_d;
}

// x_out[dst[e],f] += sum_t scores[batch[e],t] * out_all[t][e,f]
__global__ void final_scatter(const float* __restrict__ out_all,
                              const float* __restrict__ scores,
                              const int* __restrict__ batch,
                              const int* __restrict__ dst,
                              float* __restrict__ x_out)
{
    long idx = (long)blockIdx.x * blockDim.x + threadIdx.x;
    if (idx >= (long)NE * NF) return;
    const int e = (int)(idx >> 6);
    const int f = (int)(idx & 63);
    const int g = batch[e];
    const float* sc = scores + (size_t)g * NITER;
    const size_t EF = (size_t)NE * NF;
    float acc = sc[0] * out_all[idx]
              + sc[1] * out_all[idx + EF]
              + sc[2] * out_all[idx + 2 * EF]
              + sc[3] * out_all[idx + 3 * EF];
    atomicAdd(x_out + (size_t)dst[e] * NF + f, acc);
}

// ---------------------------------------------------------------------------

extern "C" void kernel_launch(void* const* d_in, const int* in_sizes, int n_in,
                              void* d_out, int out_size, void* d_ws, size_t ws_size,
                              hipStream_t stream)
{
    const float* x          = (const float*)d_in[0];
    const int*   edge_index = (const int*)  d_in[1];   // [2, NE]
    const float* edge_attr  = (const float*)d_in[2];   // [NE, 16]
    const int*   lg         = (const int*)  d_in[3];   // [2, NLG]
    const int*   batch      = (const int*)  d_in[4];   // [NE] sorted
    const float* mlp_w      = (const float*)d_in[5];
    const float* mlp_b      = (const float*)d_in[6];
    const float* lin_u_w    = (const float*)d_in[7];
    const float* lin_v_w    = (const float*)d_in[8];
    const float* lin_edge_w = (const float*)d_in[9];
    const float* conv_rel_w = (const float*)d_in[10];
    const float* conv_rel_b = (const float*)d_in[11];
    const float* conv_root_w= (const float*)d_in[12];
    const float* lin_gout_w = (const float*)d_in[13];
    const float* lin_gout_b = (const float*)d_in[14];
    const float* att_w      = (const float*)d_in[15];
    float* x_out = (float*)d_out;

    const int* src  = edge_index;            // edge_index[0]
    const int* dst  = edge_index + NE;       // edge_index[1]
    const int* lsrc = lg;                    // line_graph_edge_index[0]
    const int* ldst = lg + NLG;              // line_graph_edge_index[1]

    // workspace layout (floats)
    float* ws       = (float*)d_ws;
    float* x_in     = ws;                                    // NN*NF
    float* edge_u   = x_in   + (size_t)NN * NF;              // NN*NF
    float* edge_v   = edge_u + (size_t)NN * NF;              // NN*NF
    float* e0       = edge_v + (size_t)NN * NF;              // NE*NF
    float* out_all  = e0     + (size_t)NE * NF;              // NITER*NE*NF
    float* rbuf     = out_all+ (size_t)NITER * NE * NF;      // NE
    float* xconv    = rbuf   + NE;                           // NE
    float* gx       = xconv  + NE;                           // NG*NF
    float* gout_all = gx     + (size_t)NG * NF;              // NITER*NG*NF
    float* scores   = gout_all + (size_t)NITER * NG * NF;    // NG*NITER

    const long EF  = (long)NE * NF;
    const int  TPB = 256;
    const int  gEF  = (int)((EF + TPB - 1) / TPB);
    const int  gLGF = (int)(((long)NLG * NF + TPB - 1) / TPB);

    // 1) x_in = leaky_relu(x @ mlp_w + mlp_b)
    gemm_wmma_f32<<<NN / 16, 32, 0, stream>>>(x, NF, NF, mlp_w, mlp_b, x_in, ACT_LRELU);
    // 2) edge_u / edge_v = x_in @ lin_{u,v}_w
    gemm_wmma_f32<<<NN / 16, 32, 0, stream>>>(x_in, NF, NF, lin_u_w, nullptr, edge_u, ACT_NONE);
    gemm_wmma_f32<<<NN / 16, 32, 0, stream>>>(x_in, NF, NF, lin_v_w, nullptr, edge_v, ACT_NONE);
    // 3) e0 <- edge_attr @ lin_edge_w, then combine gathers in place
    gemm_wmma_f32<<<NE / 16, 32, 0, stream>>>(edge_attr, EDIM, EDIM, lin_edge_w, nullptr, e0, ACT_NONE);
    e0_combine<<<gEF, TPB, 0, stream>>>(edge_u, edge_v, src, dst, e0);

    // 4) message-passing iterations
    for (int t = 0; t < NITER; t++) {
        float* out_t = out_all + (size_t)t * EF;
        const float* prev = (t == 0) ? e0 : out_all + (size_t)(t - 1) * EF;

        copy_f<<<gEF, TPB, 0, stream>>>(out_t, e0, EF);
        lg_scatter_feat<<<gLGF, TPB, 0, stream>>>(lsrc, ldst, prev, out_t);

        edge_dots<<<(NE + 7) / 8, TPB, 0, stream>>>(out_t, conv_rel_w, conv_root_w,
                                                    conv_rel_b, rbuf, xconv);
        lg_scatter_scalar<<<(NLG + TPB - 1) / TPB, TPB, 0, stream>>>(lsrc, ldst, rbuf, xconv);

        graph_pool<<<NG, TPB, 0, stream>>>(xconv, batch, out_t, gx);

        gemm_wmma_f32<<<NG / 16, 32, 0, stream>>>(gx, NF, NF, lin_gout_w, lin_gout_b,
                                                  gout_all + (size_t)t * NG * NF, ACT_TANH);
    }

    // 5) attention over iterations + final node scatter
    scores_kernel<<<(NG + TPB - 1) / TPB, TPB, 0, stream>>>(gout_all, att_w, scores);
    copy_f<<<(int)(((long)NN * NF + TPB - 1) / TPB), TPB, 0, stream>>>(x_out, x_in, (long)NN * NF);
    final_scatter<<<gEF, TPB, 0, stream>>>(out_all, scores, batch, dst, x_out);
}